// ScorePredictor_2267742732805
// MI455X (gfx1250) — compile-verified
//
#include <hip/hip_runtime.h>

// ScorePredictor on gfx1250 (MI455X), compile-only tuned.
//
// score[i] = dot(x[src[i]],W0) + dot(x[dst[i]],W1) + dot(e[i],W2) + b
//
// Two-phase, bandwidth-optimal formulation:
//   Phase 1 (node kernel): sxdx[n] = { x[n].W0 , x[n].W1 }  (reads x once, 102 MB)
//   Phase 2 (edge kernel): out[i]  = e[i].W2 + sxdx[src].x + sxdx[dst].y + b
//                                                       (reads e once, 328 MB)
// Total ~432 MB @ 23.3 TB/s ~ 18.5 us, vs 983 MB naive edge-gather.
// Dots are mapped on V_WMMA_F32_16X16X4_F32 (exact f32 FMA, K accumulated 128x).

typedef float v2f __attribute__((ext_vector_type(2)));
typedef float v8f __attribute__((ext_vector_type(8)));

#define DF 512  // feature dim

// ---------------------------------------------------------------------------
// Phase 1: per-node partial scores. One wave handles a 16-node tile.
// A(16x4 f32): lane m=lane&15 is row M=m; khalf=lane>>4 selects K pair {0,1}/{2,3}
//   -> contiguous float2 load at xrow[k0 + 2*khalf].
// B(4x16 f32): lane's column n=lane&15; only n=0 (W0) and n=1 (W1) are nonzero,
//   zeroed via multiply so EXEC stays all-1s for every WMMA.
// D(16x16 f32): column n lives in lanes n / n+16, VGPR r -> row r / r+8.
// ---------------------------------------------------------------------------
__global__ __launch_bounds__(128) void score_node_partials_wmma(
    const float* __restrict__ x, const float* __restrict__ W,
    float* __restrict__ sxdx, int nNodes) {
  const int lane = threadIdx.x & 31;
  const int wave = threadIdx.x >> 5;
  const int tileBase = (blockIdx.x * 4 + wave) * 16;
  if (tileBase >= nNodes) return;  // wave-uniform exit: EXEC stays full below

  const int m = lane & 15;      // A row / B column owned by this lane
  const int khalf = lane >> 4;  // K sub-pair: 0 -> K{0,1}, 1 -> K{2,3}
  int node = tileBase + m;
  if (node >= nNodes) node = nNodes - 1;  // clamp loads; stores are predicated
  const float* __restrict__ xrow = x + (size_t)node * DF;

  // Column 0 <- W0 (src weights), column 1 <- W1 (dst weights), others zero.
  const float* __restrict__ wrow = W + ((m < 2) ? m * DF : 0);
  const float wscale = (m < 2) ? 1.0f : 0.0f;

  v8f acc = {};
#pragma unroll 4
  for (int k0 = 0; k0 < DF; k0 += 4) {
    const int kk = k0 + 2 * khalf;
    v2f A = *(const v2f*)(xrow + kk);           // global_load_b64, coalesced
    v2f B;
    B.x = wrow[kk] * wscale;                    // L0-resident weight reads
    B.y = wrow[kk + 1] * wscale;
    acc = __builtin_amdgcn_wmma_f32_16x16x4_f32(
        /*neg_a=*/false, A, /*neg_b=*/false, B,
        /*c_mod=*/(short)0, acc, /*reuse_a=*/false, /*reuse_b=*/false);
  }

  // Writeout: lanes with n<2 hold the useful columns.
  if (m < 2) {
    const int mbase = khalf * 8;
#pragma unroll
    for (int r = 0; r < 8; ++r) {
      const int nd = tileBase + mbase + r;
      if (nd < nNodes) sxdx[nd * 2 + m] = acc[r];
    }
  }
}

// ---------------------------------------------------------------------------
// Phase 2: per-edge score. One wave handles a 16-edge tile; same WMMA shape,
// only column 0 (W2) used. Lanes 0 / 16 own the dot results and fold in the
// tiny sx/dx gathers + bias (2 lanes x 8 rows of scalar traffic — negligible).
// ---------------------------------------------------------------------------
__global__ __launch_bounds__(128) void score_edge_wmma(
    const float* __restrict__ e, const float* __restrict__ We,
    const int* __restrict__ src, const int* __restrict__ dst,
    const float* __restrict__ sxdx, const float* __restrict__ bptr,
    float* __restrict__ out, int nEdges) {
  const int lane = threadIdx.x & 31;
  const int wave = threadIdx.x >> 5;
  const int tileBase = (blockIdx.x * 4 + wave) * 16;
  if (tileBase >= nEdges) return;  // wave-uniform

  const int m = lane & 15;
  const int khalf = lane >> 4;
  int edge = tileBase + m;
  if (edge >= nEdges) edge = nEdges - 1;
  const float* __restrict__ erow = e + (size_t)edge * DF;
  const float wscale = (m == 0) ? 1.0f : 0.0f;  // only column 0 carries W2

  v8f acc = {};
#pragma unroll 4
  for (int k0 = 0; k0 < DF; k0 += 4) {
    const int kk = k0 + 2 * khalf;
    v2f A = *(const v2f*)(erow + kk);
    v2f B;
    B.x = We[kk] * wscale;
    B.y = We[kk + 1] * wscale;
    acc = __builtin_amdgcn_wmma_f32_16x16x4_f32(
        false, A, false, B, (short)0, acc, false, false);
  }

  if (m == 0) {  // lanes 0 and 16 hold column 0 for rows 0..7 / 8..15
    const float bias = *bptr;
    const int mbase = khalf * 8;
#pragma unroll
    for (int r = 0; r < 8; ++r) {
      const int ei = tileBase + mbase + r;
      if (ei < nEdges) {
        const int s = src[ei];
        const int d = dst[ei];
        out[ei] = acc[r] + sxdx[2 * s] + sxdx[2 * d + 1] + bias;
      }
    }
  }
}

extern "C" void kernel_launch(void* const* d_in, const int* in_sizes, int n_in,
                              void* d_out, int out_size, void* d_ws, size_t ws_size,
                              hipStream_t stream) {
  const float* x   = (const float*)d_in[0];  // [N_NODES, 512]
  const float* e   = (const float*)d_in[1];  // [N_EDGES, 512]
  const int*   src = (const int*)d_in[2];    // [N_EDGES]
  const int*   dst = (const int*)d_in[3];    // [N_EDGES]
  const float* W   = (const float*)d_in[4];  // [1, 1536] = [W0 | W1 | W2]
  const float* b   = (const float*)d_in[5];  // [1]
  float* out = (float*)d_out;                // [N_EDGES]

  const int nNodes = in_sizes[0] / DF;
  const int nEdges = in_sizes[2];
  float* sxdx = (float*)d_ws;  // [nNodes][2] f32 partials (400 KB)

  const int nodeTiles = (nNodes + 15) / 16;
  const int nodeBlocks = (nodeTiles + 3) / 4;  // 4 waves (16-row tiles) / block
  score_node_partials_wmma<<<nodeBlocks, 128, 0, stream>>>(x, W, sxdx, nNodes);

  const int edgeTiles = (nEdges + 15) / 16;
  const int edgeBlocks = (edgeTiles + 3) / 4;
  score_edge_wmma<<<edgeBlocks, 128, 0, stream>>>(e, W + 2 * DF, src, dst, sxdx,
                                                  b, out, nEdges);
}